// AMU_77309411328339
// MI455X (gfx1250) — compile-verified
//
#include <hip/hip_runtime.h>
#include <math.h>

// B=32, T=256, IN=512, D=64, NB=128, N1=129, OUT=512
// Padded column stride for N1=129 projections: 144 (multiple of 16)

typedef float v2f __attribute__((ext_vector_type(2)));
typedef float v8f __attribute__((ext_vector_type(8)));

// ---------------------------------------------------------------------------
// Zero-pad a (rows x N) row-major matrix into (rows x Npad); total = rows*Npad.
// Also used to produce zero buffers (N = 0).
// ---------------------------------------------------------------------------
__global__ void pad_weight(const float* __restrict__ src, float* __restrict__ dst,
                           int N, int Npad, int total)
{
    const int i = blockIdx.x * blockDim.x + threadIdx.x;
    if (i >= total) return;
    const int row = i / Npad;
    const int col = i - row * Npad;
    dst[i] = (col < N) ? src[row * N + col] : 0.0f;
}

// ---------------------------------------------------------------------------
// Generic strided-batched f32 WMMA GEMM: C[z] = alpha * A[z] @ op(B[z]) + bias
// One wave computes one 16x16 C tile with V_WMMA_F32_16X16X4_F32.
// blockDim.x = 128 (4 waves); M-tile = blockIdx.x*4+wave; N-tile = blockIdx.y;
// batch z = blockIdx.z.
// Caller guarantees: M == 64*gridDim.x, K % 4 == 0, every B element addressed
// for n < 16*gridDim.y is in-range (zero-padded), bias has 16*gridDim.y
// entries (zero-padded). Branch-free inner loop; A and transposed-B use
// 8-byte vector loads (all strides even).
// ---------------------------------------------------------------------------
template<bool TRANSB>
__global__ void wmma_gemm(const float* __restrict__ A, int lda, long sA,
                          const float* __restrict__ Bm, int ldb, long sB,
                          float* __restrict__ C, int ldc, long sC,
                          const float* __restrict__ bias, int relu,
                          int K, float alpha)
{
    const int wave = threadIdx.x >> 5;
    const int lane = threadIdx.x & 31;
    const int mt = blockIdx.x * (blockDim.x >> 5) + wave;
    const int nt = blockIdx.y;
    const long z = blockIdx.z;

    const float* __restrict__ Ab = A + z * sA;
    const float* __restrict__ Bb = Bm + z * sB;
    float* __restrict__ Cb = C + z * sC;

    const int half = lane >> 4;      // 0: K pair {0,1}; 1: K pair {2,3}
    const int r = lane & 15;         // M index (A) / N index (B,C)
    const long m = mt * 16 + r;
    const int n = nt * 16 + r;

    v8f acc = {};
#pragma unroll 4
    for (int kk = 0; kk < K; kk += 4) {
        const int k0 = kk + 2 * half;
        const v2f a = *(const v2f*)(Ab + m * (long)lda + k0);
        v2f b;
        if (TRANSB) {
            b = *(const v2f*)(Bb + (long)n * ldb + k0);
        } else {
            b[0] = Bb[(long)k0 * ldb + n];
            b[1] = Bb[(long)(k0 + 1) * ldb + n];
        }
        acc = __builtin_amdgcn_wmma_f32_16x16x4_f32(
            false, a, false, b, (short)0, acc, false, false);
    }

    // C layout: VGPR r2 -> row r2 + 8*half; lane&15 -> column
    const float bv = bias[n];
#pragma unroll
    for (int r2 = 0; r2 < 8; ++r2) {
        float v = acc[r2] * alpha + bv;
        if (relu) v = fmaxf(v, 0.0f);
        Cb[(long)(mt * 16 + r2 + 8 * half) * ldc + n] = v;
    }
}

// ---------------------------------------------------------------------------
// Softmax over the time axis: for each (b, n) column of scores (B,T,stride144)
// in-place; also emit attn[b*144+n] = sw[b, T-1, n]  (attn stride 144).
// ---------------------------------------------------------------------------
__global__ void softmax_time(float* __restrict__ sw, float* __restrict__ attn)
{
    const int b = blockIdx.x;
    const int n = blockIdx.y * 64 + threadIdx.x;
    if (n >= 129) return;
    float* col = sw + (long)b * 256 * 144 + n;
    float mx = -INFINITY;
    for (int t = 0; t < 256; ++t) mx = fmaxf(mx, col[t * 144]);
    float s = 0.0f;
    for (int t = 0; t < 256; ++t) s += __expf(col[t * 144] - mx);
    const float inv = 1.0f / s;
    float last = 0.0f;
    for (int t = 0; t < 256; ++t) {
        const float v = __expf(col[t * 144] - mx) * inv;
        col[t * 144] = v;
        last = v;
    }
    attn[b * 144 + n] = last;
}

// ---------------------------------------------------------------------------
// weakest[b,t] = first argmin over n<129 of sw[b,t,n]
// ---------------------------------------------------------------------------
__global__ void argmin_rows(const float* __restrict__ sw, int* __restrict__ weakest)
{
    const int b = blockIdx.x;
    const int t = threadIdx.x;
    const float* row = sw + (long)b * 256 * 144 + (long)t * 144;
    float best = row[0];
    int bi = 0;
    for (int n = 1; n < 129; ++n) {
        const float v = row[n];
        if (v < best) { best = v; bi = n; }
    }
    weakest[b * 256 + t] = bi;
}

// ---------------------------------------------------------------------------
// Sequential memory scan. The torch scatter writes identical (slot,value)
// pairs into every batch and the initial memory is batch-uniform, so the
// carried memory is (NB,D) shared across batches. One block, 256 threads,
// state in LDS (40 KB of the 320 KB WGP pool).
// Emits: Mhist (T,NB,D) shared post-scatter memory, Lbuf (T,B,D) last slot,
// and final memory replicated to d_out's second region (B,NB,D).
// ---------------------------------------------------------------------------
__global__ void scan_kernel(const float* __restrict__ entry,   // (B,T,D)
                            const float* __restrict__ mem_in,  // (B,NB,D); batch 0
                            const int* __restrict__ weakest,   // (B,T)
                            float* __restrict__ Mhist,         // (T,NB,D)
                            float* __restrict__ Lbuf,          // (T,B,D)
                            float* __restrict__ final_out)     // (B,NB,D)
{
    __shared__ float mem[128 * 64];
    __shared__ float ebuf[32 * 64];
    __shared__ int lastw[129];
    __shared__ int wv[32];
    const int tid = threadIdx.x;

    for (int i = tid; i < 8192; i += 256) mem[i] = mem_in[i];
    __syncthreads();

    for (int t = 0; t < 256; ++t) {
        for (int i = tid; i < 2048; i += 256) {
            const int j = i >> 6, d = i & 63;
            ebuf[i] = entry[((long)j * 256 + t) * 64 + d];
        }
        if (tid < 32) wv[tid] = weakest[tid * 256 + t];
        __syncthreads();
        if (tid < 129) {
            int lw = -1;
            for (int j = 0; j < 32; ++j)
                if (wv[j] == tid) lw = j;   // last writer wins
            lastw[tid] = lw;
        }
        __syncthreads();
        for (int i = tid; i < 8192; i += 256) {
            const int n = i >> 6, d = i & 63;
            const int lw = lastw[n];
            if (lw >= 0) mem[i] = ebuf[lw * 64 + d];
        }
        __syncthreads();
        for (int i = tid; i < 8192; i += 256)
            Mhist[(long)t * 8192 + i] = mem[i];
        const int lN = lastw[128];
        for (int i = tid; i < 2048; i += 256) {
            const int d = i & 63;
            Lbuf[(long)t * 2048 + i] = (lN >= 0) ? ebuf[lN * 64 + d] : ebuf[i];
        }
        __syncthreads();
    }
    for (int b = 0; b < 32; ++b)
        for (int i = tid; i < 8192; i += 256)
            final_out[(long)b * 8192 + i] = mem[i];
}

// ---------------------------------------------------------------------------
// Final per-t GEMM with the attention weights and per-batch last slot folded
// into a K=192 contraction, split at the wave-uniform k=128 boundary:
//   out[t*32+b, o] = sum_{k<128} attn[b,k]*Y[t,k,o]
//                  + attn[b,128] * sum_{d<64} L[t,b,d]*Wl[8192+d, o] + bl[o]
// M=32 (2 waves), N=512 (grid.y=32), batch = t (grid.z=256).
// ---------------------------------------------------------------------------
__global__ void final_gemm(const float* __restrict__ attn,  // (B, stride 144)
                           const float* __restrict__ Lbuf,  // (T,B,64)
                           const float* __restrict__ Y,     // (T,128,512)
                           const float* __restrict__ Wl,    // (8256,512)
                           const float* __restrict__ bl,
                           float* __restrict__ out)         // rows r=t*32+b
{
    const int wave = threadIdx.x >> 5;
    const int lane = threadIdx.x & 31;
    const int half = lane >> 4;
    const int r = lane & 15;
    const int t = blockIdx.z;
    const int nt = blockIdx.y;
    const int b_base = wave * 16;

    const float* __restrict__ Yt = Y + (long)t * 128 * 512;
    const float* __restrict__ Lt = Lbuf + (long)t * 32 * 64;

    const int brow = b_base + r;
    const int ncol = nt * 16 + r;

    v8f acc = {};
#pragma unroll 4
    for (int kk = 0; kk < 128; kk += 4) {
        const int k0 = kk + 2 * half;
        const v2f a = *(const v2f*)(attn + brow * 144 + k0);
        v2f b;
        b[0] = Yt[(long)k0 * 512 + ncol];
        b[1] = Yt[(long)(k0 + 1) * 512 + ncol];
        acc = __builtin_amdgcn_wmma_f32_16x16x4_f32(
            false, a, false, b, (short)0, acc, false, false);
    }
    const float aN = attn[brow * 144 + 128];
#pragma unroll 4
    for (int kk = 0; kk < 64; kk += 4) {
        const int k0 = kk + 2 * half;
        const v2f l = *(const v2f*)(Lt + brow * 64 + k0);
        v2f a;
        a[0] = aN * l[0];
        a[1] = aN * l[1];
        v2f b;
        b[0] = Wl[(long)(8192 + k0) * 512 + ncol];
        b[1] = Wl[(long)(8193 + k0) * 512 + ncol];
        acc = __builtin_amdgcn_wmma_f32_16x16x4_f32(
            false, a, false, b, (short)0, acc, false, false);
    }

    const float bias = bl[ncol];
#pragma unroll
    for (int r2 = 0; r2 < 8; ++r2) {
        const int bb = b_base + r2 + 8 * half;
        const long rowr = (long)t * 32 + bb;
        out[rowr * 512 + ncol] = acc[r2] + bias;
    }
}

// ---------------------------------------------------------------------------
extern "C" void kernel_launch(void* const* d_in, const int* in_sizes, int n_in,
                              void* d_out, int out_size, void* d_ws, size_t ws_size,
                              hipStream_t stream)
{
    const float* inp    = (const float*)d_in[0];
    const float* memory = (const float*)d_in[1];
    const float* Wq     = (const float*)d_in[2];
    const float* bq     = (const float*)d_in[3];
    const float* Wk     = (const float*)d_in[4];
    const float* bk     = (const float*)d_in[5];
    const float* Wa     = (const float*)d_in[6];
    const float* ba     = (const float*)d_in[7];
    const float* We     = (const float*)d_in[8];
    const float* be     = (const float*)d_in[9];
    const float* Wl     = (const float*)d_in[10];
    const float* bl     = (const float*)d_in[11];
    float* out = (float*)d_out;

    // Workspace layout (floats)
    float* qb   = (float*)d_ws;          // 8192 x 144
    float* kb   = qb + 1179648;          // 8192 x 144
    float* mab  = kb + 1179648;          // 8192 x 144
    float* eb   = mab + 1179648;         // 8192 x 64
    float* s1   = eb + 524288;           // 32 x 256 x 256
    float* sw   = s1 + 2097152;          // 32 x 256 x 144
    float* attn = sw + 1179648;          // 32 x 144
    float* Mh   = attn + 4608;           // 256 x 128 x 64
    float* Lb   = Mh + 2097152;          // 256 x 32 x 64
    float* Yb   = Lb + 524288;           // 256 x 128 x 512
    float* Wqp  = Yb + 16777216;         // 512 x 144 zero-padded
    float* Wkp  = Wqp + 73728;           // 512 x 144
    float* Wap  = Wkp + 73728;           // 512 x 144
    float* bqp  = Wap + 73728;           // 144
    float* bkp  = bqp + 144;             // 144
    float* bap  = bkp + 144;             // 144
    float* zb   = bap + 144;             // 512 zeros
    int*   wk   = (int*)(zb + 512);      // 32 x 256

    // --- zero-pad weights/biases so GEMM inner loops need no bounds checks ---
    pad_weight<<<(73728 + 255) / 256, 256, 0, stream>>>(Wq, Wqp, 129, 144, 73728);
    pad_weight<<<(73728 + 255) / 256, 256, 0, stream>>>(Wk, Wkp, 129, 144, 73728);
    pad_weight<<<(73728 + 255) / 256, 256, 0, stream>>>(Wa, Wap, 129, 144, 73728);
    pad_weight<<<1, 144, 0, stream>>>(bq, bqp, 129, 144, 144);
    pad_weight<<<1, 144, 0, stream>>>(bk, bkp, 129, 144, 144);
    pad_weight<<<1, 144, 0, stream>>>(ba, bap, 129, 144, 144);
    pad_weight<<<2, 256, 0, stream>>>(bq, zb, 0, 512, 512);   // zeros

    const dim3 blk(128);

    // Projections: q, k, ma (padded 512x144 weights), entry (relu)
    wmma_gemm<false><<<dim3(128, 9, 1), blk, 0, stream>>>(
        inp, 512, 0, Wqp, 144, 0, qb, 144, 0, bqp, 0, 512, 1.0f);
    wmma_gemm<false><<<dim3(128, 9, 1), blk, 0, stream>>>(
        inp, 512, 0, Wkp, 144, 0, kb, 144, 0, bkp, 0, 512, 1.0f);
    wmma_gemm<false><<<dim3(128, 9, 1), blk, 0, stream>>>(
        inp, 512, 0, Wap, 144, 0, mab, 144, 0, bap, 0, 512, 1.0f);
    wmma_gemm<false><<<dim3(128, 4, 1), blk, 0, stream>>>(
        inp, 512, 0, We, 64, 0, eb, 64, 0, be, 1, 512, 1.0f);

    // scores1[b] = (q_b @ k_b^T) / sqrt(129): K=144 (pads are zero)
    const float inv = 1.0f / sqrtf(129.0f);
    wmma_gemm<true><<<dim3(4, 16, 32), blk, 0, stream>>>(
        qb, 144, 36864, kb, 144, 36864, s1, 256, 65536, zb, 0, 144, inv);

    // scores2[b] = scores1[b] @ ma_b : (256x256)@(256x144pad) -> (256x144)
    wmma_gemm<false><<<dim3(4, 9, 32), blk, 0, stream>>>(
        s1, 256, 65536, mab, 144, 36864, sw, 144, 36864, zb, 0, 256, 1.0f);

    softmax_time<<<dim3(32, 3), 64, 0, stream>>>(sw, attn);
    argmin_rows<<<32, 256, 0, stream>>>(sw, wk);

    // Sequential scan; also writes final_mem region of d_out (offset B*T*OUT)
    scan_kernel<<<1, 256, 0, stream>>>(eb, memory, wk, Mh, Lb, out + 4194304);

    // Y[t,n,:] = Mhist[t,n,:] @ Wl[n*64:(n+1)*64, :]  (128 block-diag GEMMs, z=n)
    wmma_gemm<false><<<dim3(4, 32, 128), blk, 0, stream>>>(
        Mh, 8192, 64, Wl, 512, 32768, Yb, 65536, 512, zb, 0, 64, 1.0f);

    // Final: per-t (32 x 512, K=192) GEMM with attention + last-slot folded in
    final_gemm<<<dim3(1, 32, 256), 64, 0, stream>>>(attn, Lb, Yb, Wl, bl, out);
}